// MyLLMMoERouter_55250459295816
// MI455X (gfx1250) — compile-verified
//
#include <hip/hip_runtime.h>
#include <cmath>

typedef __attribute__((ext_vector_type(2))) float v2f;
typedef __attribute__((ext_vector_type(8))) float v8f;

static constexpr int HID       = 2048;
static constexpr int NE        = 64;
static constexpr int TOPK      = 8;
static constexpr int NTOK      = 4 * 4096;   // bsz*seq = 16384 tokens
static constexpr int WG_TOKENS = 128;        // tokens per workgroup (8 waves x 16)
static constexpr int LDS_STRIDE = NE + 1;    // 65: odd stride -> no LDS bank conflicts

__global__ __launch_bounds__(256)
void moe_router_gate_topk(const float* __restrict__ x,
                          const float* __restrict__ W,
                          const float* __restrict__ bias,
                          float* __restrict__ out) {
  __shared__ float ldsGate[WG_TOKENS * LDS_STRIDE];  // 33.3 KB
  __shared__ float ldsIds [WG_TOKENS * TOPK];        // 4 KB
  __shared__ float ldsBias[NE];

  const int tid    = threadIdx.x;
  const int wave   = tid >> 5;       // 0..7 (wave32)
  const int lane   = tid & 31;
  const int l15    = lane & 15;
  const int lhalf  = lane >> 4;      // 0 or 1
  const int wgTok  = blockIdx.x * WG_TOKENS;
  const int waveTok = wgTok + wave * 16;

  if (tid < NE) ldsBias[tid] = bias[tid];

  // ---- GEMM: 16 tokens x 64 experts per wave, K = 2048 in chunks of 4 ----
  // A (16x4 f32): lane l15 = row M; lanes 0-15 hold K={k,k+1}, lanes 16-31 K={k+2,k+3}
  // B (4x16 f32): lane l15 = col N (= expert); same K split across lane halves
  const float* xP  = x + (size_t)(waveTok + l15) * HID + (lhalf << 1);
  const float* wP0 = W + (size_t)(l15) * HID + (lhalf << 1);
  const float* wP1 = wP0 + 16 * HID;
  const float* wP2 = wP0 + 32 * HID;
  const float* wP3 = wP0 + 48 * HID;

  v8f acc0 = {}; v8f acc1 = {}; v8f acc2 = {}; v8f acc3 = {};

  #pragma unroll 4
  for (int kk = 0; kk < HID; kk += 4) {
    v2f a  = *(const v2f*)(xP  + kk);
    v2f b0 = *(const v2f*)(wP0 + kk);
    v2f b1 = *(const v2f*)(wP1 + kk);
    v2f b2 = *(const v2f*)(wP2 + kk);
    v2f b3 = *(const v2f*)(wP3 + kk);
    acc0 = __builtin_amdgcn_wmma_f32_16x16x4_f32(false, a, false, b0, (short)0, acc0, false, false);
    acc1 = __builtin_amdgcn_wmma_f32_16x16x4_f32(false, a, false, b1, (short)0, acc1, false, false);
    acc2 = __builtin_amdgcn_wmma_f32_16x16x4_f32(false, a, false, b2, (short)0, acc2, false, false);
    acc3 = __builtin_amdgcn_wmma_f32_16x16x4_f32(false, a, false, b3, (short)0, acc3, false, false);
  }

  // ---- Scatter C tiles to LDS (token-major, padded stride) ----
  // C/D layout: VGPR r, lanes 0-15 -> M=r, lanes 16-31 -> M=r+8; N = l15
  const int trow = wave * 16 + 8 * lhalf;
  #pragma unroll
  for (int r = 0; r < 8; ++r) {
    ldsGate[(trow + r) * LDS_STRIDE +  0 + l15] = acc0[r];
    ldsGate[(trow + r) * LDS_STRIDE + 16 + l15] = acc1[r];
    ldsGate[(trow + r) * LDS_STRIDE + 32 + l15] = acc2[r];
    ldsGate[(trow + r) * LDS_STRIDE + 48 + l15] = acc3[r];
  }
  __syncthreads();

  // ---- Per-token top-8 + masked softmax (one thread per token) ----
  if (tid < WG_TOKENS) {
    float* g = &ldsGate[tid * LDS_STRIDE];
    #pragma unroll
    for (int e = 0; e < NE; ++e) g[e] += ldsBias[e];

    float vals[TOPK];
    int   ids [TOPK];
    unsigned long long used = 0ull;
    #pragma unroll
    for (int j = 0; j < TOPK; ++j) {
      float best = -INFINITY;
      int   bi   = 0;
      #pragma unroll 8
      for (int e = 0; e < NE; ++e) {
        float v = g[e];
        bool ok = (((used >> e) & 1ull) == 0ull) && (v > best);  // strict > : lowest-index tiebreak
        best = ok ? v : best;
        bi   = ok ? e : bi;
      }
      vals[j] = best;
      ids[j]  = bi;
      used |= (1ull << bi);
    }

    const float mx = vals[0];               // global max (top-1)
    float ex[TOPK];
    float sum = 0.f;
    #pragma unroll
    for (int j = 0; j < TOPK; ++j) { ex[j] = __expf(vals[j] - mx); sum += ex[j]; }
    const float inv = 1.f / sum;

    // overwrite this token's gate row with probabilities (zeros outside top-k)
    #pragma unroll
    for (int e = 0; e < NE; ++e) g[e] = 0.f;
    #pragma unroll
    for (int j = 0; j < TOPK; ++j) g[ids[j]] = ex[j] * inv;
    #pragma unroll
    for (int j = 0; j < TOPK; ++j) ldsIds[tid * TOPK + j] = (float)ids[j];
  }
  __syncthreads();

  // ---- Coalesced writeback: probs then ids (concatenated flat outputs) ----
  float* probsOut = out;
  float* idsOut   = out + (size_t)NTOK * NE;
  for (int i = tid; i < WG_TOKENS * NE; i += 256) {
    const int t = i >> 6;
    const int e = i & 63;
    probsOut[(size_t)wgTok * NE + i] = ldsGate[t * LDS_STRIDE + e];
  }
  for (int i = tid; i < WG_TOKENS * TOPK; i += 256) {
    idsOut[(size_t)wgTok * TOPK + i] = ldsIds[i];
  }
}

extern "C" void kernel_launch(void* const* d_in, const int* in_sizes, int n_in,
                              void* d_out, int out_size, void* d_ws, size_t ws_size,
                              hipStream_t stream) {
  const float* x = (const float*)d_in[0];   // (4, 4096, 2048) f32
  const float* W = (const float*)d_in[1];   // (64, 2048) f32
  const float* b = (const float*)d_in[2];   // (64,) f32
  float* out = (float*)d_out;               // probs (16384*64) ++ ids (16384*8)

  dim3 grid(NTOK / WG_TOKENS);              // 128 workgroups
  dim3 block(256);                          // 8 wave32 waves
  moe_router_gate_topk<<<grid, block, 0, stream>>>(x, W, b, out);
}